// DifferentiableSuperpixelTokenizer_72894184947795
// MI455X (gfx1250) — compile-verified
//
#include <hip/hip_runtime.h>

typedef __attribute__((ext_vector_type(2))) float v2f;
typedef __attribute__((ext_vector_type(8))) float v8f;

#define B_IMG 64
#define CCH   3
#define HH    512
#define WW    512
#define NSEG  1024
#define BOX   16
#define HALF  8
#define EDIM  768
#define KDIM  768   // C*box*box = 3*16*16

// ---------------- kernel 1: zero centroid sums ----------------
__global__ __launch_bounds__(256) void zero_sums(float* __restrict__ sums) {
    int i = blockIdx.x * 256 + threadIdx.x;   // B*S*3 = 196608 total
    if (i < B_IMG * NSEG * 3) sums[i] = 0.0f;
}

// ---------------- kernel 2: pack conv_w into K-pair-major float2 ----------------
// bpack[p][e] = (conv_w[e][2p], conv_w[e][2p+1]);  p in [0,384), e in [0,768)
__global__ __launch_bounds__(256) void pack_weights(const float* __restrict__ conv_w,
                                                    float2* __restrict__ bpack) {
    int i = blockIdx.x * 256 + threadIdx.x;   // 768*384 total
    if (i >= EDIM * (KDIM / 2)) return;
    int e = i % EDIM;
    int p = i / EDIM;
    float2 v = *(const float2*)(conv_w + (size_t)e * KDIM + 2 * p);
    bpack[(size_t)p * EDIM + e] = v;
}

// ---------------- kernel 3: centroid scatter-add (LDS reduce, then global atomics) ----
__global__ __launch_bounds__(256) void centroid_scatter(const int* __restrict__ seg,
                                                        float* __restrict__ sums) {
    __shared__ float sx[NSEG];
    __shared__ float sy[NSEG];
    __shared__ float sc[NSEG];
    const int tid = threadIdx.x;
    for (int i = tid; i < NSEG; i += 256) { sx[i] = 0.f; sy[i] = 0.f; sc[i] = 0.f; }
    __syncthreads();

    const int b     = blockIdx.x >> 4;        // 16 blocks per image
    const int chunk = blockIdx.x & 15;
    const int base  = chunk * (HH * WW / 16); // 16384 pixels per block
    const int* segb = seg + (size_t)b * HH * WW;

    for (int i = tid; i < HH * WW / 16; i += 256) {
        int p = base + i;
        int s = segb[p];
        atomicAdd(&sx[s], (float)(p & (WW - 1)));
        atomicAdd(&sy[s], (float)(p >> 9));
        atomicAdd(&sc[s], 1.0f);
    }
    __syncthreads();

    float* out = sums + (size_t)b * NSEG * 3;
    for (int i = tid; i < NSEG; i += 256) {
        if (sc[i] != 0.0f) {
            atomicAdd(&out[i * 3 + 0], sx[i]);
            atomicAdd(&out[i * 3 + 1], sy[i]);
            atomicAdd(&out[i * 3 + 2], sc[i]);
        }
    }
}

// ---------------- kernel 4: per-(b,s) patch origin ----------------
__global__ __launch_bounds__(256) void centroid_mins(const float* __restrict__ sums,
                                                     int* __restrict__ mins) {
    int i = blockIdx.x * 256 + threadIdx.x;   // B*S
    if (i >= B_IMG * NSEG) return;
    float cnt = fmaxf(sums[i * 3 + 2], 1e-6f);
    mins[i * 2 + 0] = (int)floorf(sums[i * 3 + 1] / cnt);  // y_min
    mins[i * 2 + 1] = (int)floorf(sums[i * 3 + 0] / cnt);  // x_min
}

// ---------------- kernel 5: gather patches -> LDS, WMMA f32 GEMM ----------------
#define SEGS_PB    32
#define LDS_STRIDE 772   // 768 + 4; stride % 64banks == 4 -> conflict-free A-frag b64 loads

__global__ __launch_bounds__(256) void tok_gemm(const float* __restrict__ img,
                                                const int* __restrict__ mins,
                                                const float2* __restrict__ bpack,
                                                float* __restrict__ out) {
    __shared__ float ldsA[SEGS_PB * LDS_STRIDE];  // 32 segments x 768 K (f32)
    __shared__ int   ldsYX[SEGS_PB * 2];

    const int b       = blockIdx.x >> 5;          // 32 segment-tiles per image
    const int segBase = (blockIdx.x & 31) * SEGS_PB;
    const int tid     = threadIdx.x;

    if (tid < SEGS_PB * 2)
        ldsYX[tid] = mins[((size_t)b * NSEG + segBase) * 2 + tid];
    __syncthreads();

    // Gather 32 patches (bounds-checked zero padding), 96 elements/thread, coalesced in k.
    for (int i = tid; i < SEGS_PB * KDIM; i += 256) {
        int si  = i / KDIM;
        int k   = i - si * KDIM;
        int c   = k >> 8;
        int rem = k & 255;
        int y   = ldsYX[si * 2 + 0] + (rem >> 4) - HALF;
        int x   = ldsYX[si * 2 + 1] + (rem & 15) - HALF;
        float v = 0.0f;
        if ((unsigned)y < (unsigned)HH && (unsigned)x < (unsigned)WW)
            v = img[(((size_t)b * CCH + c) * HH + y) * WW + x];
        ldsA[si * LDS_STRIDE + k] = v;
    }
    __syncthreads();

    const int lane  = tid & 31;
    const int wv    = tid >> 5;      // 8 waves
    const int lrow  = lane & 15;     // M within tile (A) / N within tile (B,C)
    const int khalf = lane >> 4;     // selects K pair {2*khalf, 2*khalf+1}
    const int e0    = wv * 96;       // 6 N-tiles of 16 per wave

    v8f acc[2][6];
#pragma unroll
    for (int mt = 0; mt < 2; ++mt)
#pragma unroll
        for (int j = 0; j < 6; ++j) acc[mt][j] = (v8f){0.f,0.f,0.f,0.f,0.f,0.f,0.f,0.f};

    const float* a0p = &ldsA[(0 * 16 + lrow) * LDS_STRIDE + 2 * khalf];
    const float* a1p = &ldsA[(1 * 16 + lrow) * LDS_STRIDE + 2 * khalf];

    for (int k0 = 0; k0 < KDIM; k0 += 4) {
        v2f a0 = *(const v2f*)(a0p + k0);
        v2f a1 = *(const v2f*)(a1p + k0);
        const float2* brow = bpack + (size_t)(k0 / 2 + khalf) * EDIM + e0 + lrow;
#pragma unroll
        for (int j = 0; j < 6; ++j) {
            float2 wp = brow[j * 16];
            v2f bv = {wp.x, wp.y};
            acc[0][j] = __builtin_amdgcn_wmma_f32_16x16x4_f32(
                false, a0, false, bv, (short)0, acc[0][j], false, false);
            acc[1][j] = __builtin_amdgcn_wmma_f32_16x16x4_f32(
                false, a1, false, bv, (short)0, acc[1][j], false, false);
        }
    }

    // C/D layout: lane -> N = lane%16 ; VGPR r -> M = r + 8*(lane/16)
    const size_t mBase = (size_t)b * NSEG + segBase;
#pragma unroll
    for (int mt = 0; mt < 2; ++mt)
#pragma unroll
        for (int j = 0; j < 6; ++j) {
            int ecol = e0 + j * 16 + lrow;
#pragma unroll
            for (int r = 0; r < 8; ++r) {
                int m = mt * 16 + r + 8 * khalf;
                out[(mBase + m) * EDIM + ecol] = acc[mt][j][r];
            }
        }
}

// ---------------- launch ----------------
extern "C" void kernel_launch(void* const* d_in, const int* in_sizes, int n_in,
                              void* d_out, int out_size, void* d_ws, size_t ws_size,
                              hipStream_t stream) {
    const float* img      = (const float*)d_in[0];
    const int*   segments = (const int*)d_in[1];
    const float* conv_w   = (const float*)d_in[2];
    float*       out      = (float*)d_out;

    char* ws = (char*)d_ws;
    float*  sums  = (float*)ws;                                  // B*S*3 f32   (768 KB)
    int*    mins  = (int*)(ws + (size_t)B_IMG * NSEG * 3 * 4);   // B*S*2 i32   (512 KB)
    float2* bpack = (float2*)(ws + (size_t)B_IMG * NSEG * 3 * 4
                                 + (size_t)B_IMG * NSEG * 2 * 4); // 768*768 f32 (2.25 MB)

    zero_sums<<<(B_IMG * NSEG * 3 + 255) / 256, 256, 0, stream>>>(sums);
    pack_weights<<<(EDIM * (KDIM / 2) + 255) / 256, 256, 0, stream>>>(conv_w, bpack);
    centroid_scatter<<<B_IMG * 16, 256, 0, stream>>>(segments, sums);
    centroid_mins<<<(B_IMG * NSEG + 255) / 256, 256, 0, stream>>>(sums, mins);
    tok_gemm<<<B_IMG * (NSEG / SEGS_PB), 256, 0, stream>>>(img, mins, bpack, out);
}